// GraphConvolution_50689204027734
// MI455X (gfx1250) — compile-verified
//
#include <hip/hip_runtime.h>

typedef float v2f __attribute__((ext_vector_type(2)));
typedef float v8f __attribute__((ext_vector_type(8)));

#define B_    8
#define N_    512
#define FIN   256
#define FOUT  256
#define MTOT  (B_ * N_)   // 4096 total rows

// ---------------------------------------------------------------------------
// Kernel 1: rel_mean[i,f] = (1/N) * sum_j rel_pos[i,j,f]
// One block per i (512 blocks), 256 threads.
// Thread t: subgroup g = t>>6 handles rows j = g, g+4, g+8, ... ;
// float4 column f4 = t&63 (16B loads, 1KB contiguous per 64-thread group).
// 268MB total stream -> this kernel sets the runtime (~12us at 23.3 TB/s),
// so it uses b128 loads to minimize request count.
// ---------------------------------------------------------------------------
__global__ void relmean_kernel(const float* __restrict__ rel,
                               float* __restrict__ rel_mean) {
    __shared__ float part[4][FOUT];

    const int i  = blockIdx.x;
    const int t  = threadIdx.x;
    const int g  = t >> 6;   // j subgroup: 0..3
    const int f4 = t & 63;   // float4 column: covers FOUT/4 = 64

    // base: row (j = g) of tensor slice i, column 4*f4
    const float4* p = (const float4*)(rel + (size_t)i * N_ * FOUT
                                          + (size_t)g * FOUT) + f4;
    float4 s = make_float4(0.f, 0.f, 0.f, 0.f);
#pragma unroll 8
    for (int jj = 0; jj < N_ / 4; ++jj) {
        // advance 4 rows per iteration: 4 * (FOUT/4) float4 = FOUT float4
        float4 v = p[(size_t)jj * FOUT];
        s.x += v.x; s.y += v.y; s.z += v.z; s.w += v.w;
    }
    part[g][f4 * 4 + 0] = s.x;
    part[g][f4 * 4 + 1] = s.y;
    part[g][f4 * 4 + 2] = s.z;
    part[g][f4 * 4 + 3] = s.w;
    __syncthreads();

    const float r = (part[0][t] + part[1][t] + part[2][t] + part[3][t])
                    * (1.0f / (float)N_);
    rel_mean[i * FOUT + t] = r;
}

// ---------------------------------------------------------------------------
// Kernel 2: row_sum[row] = sum_j adj[row, j], row in [0, B*N)
// 8 waves per block, one wave per row; lanes do float4 loads (4 iters),
// then wave32 butterfly reduction.
// ---------------------------------------------------------------------------
__global__ void rowsum_kernel(const float* __restrict__ adj,
                              float* __restrict__ row_sum) {
    const int row  = blockIdx.x * 8 + (threadIdx.x >> 5);
    const int lane = threadIdx.x & 31;
    const float4* p = (const float4*)(adj + (size_t)row * N_);
    float4 a = make_float4(0.f, 0.f, 0.f, 0.f);
#pragma unroll
    for (int j = lane; j < N_ / 4; j += 32) {   // 128 float4 per row -> 4 iters
        float4 v = p[j];
        a.x += v.x; a.y += v.y; a.z += v.z; a.w += v.w;
    }
    float s = (a.x + a.y) + (a.z + a.w);
#pragma unroll
    for (int off = 16; off > 0; off >>= 1) {
        s += __shfl_xor(s, off, 32);
    }
    if (lane == 0) row_sum[row] = s;
}

// ---------------------------------------------------------------------------
// Kernel 3: hidden = input @ weight via V_WMMA_F32_16X16X4_F32, fused epilogue:
//   out = relu((row_sum/N * hidden + rel_mean[i,:]) / (row_sum + 1) + bias)
//
// One wave computes one 16x16 output tile; K-loop steps by 4.
// ISA operand layouts (wave32):
//   A 16x4 : lane l -> M = l%16 ; half (l>=16) holds K+2 ; vgpr0/1 = K, K+1
//   B 4x16 : lane l -> Ncol = l%16 ; same K split as A
//   D 16x16: vgpr r, lane l -> M = r + 8*(l>=16), Ncol = l%16
// Grid: (MTOT/16, FOUT/(16*8)) blocks of 256 threads (8 waves).
// ---------------------------------------------------------------------------
__global__ void gemm_epilogue_kernel(const float* __restrict__ in,
                                     const float* __restrict__ w,
                                     const float* __restrict__ bias,
                                     const float* __restrict__ rel_mean,
                                     const float* __restrict__ row_sum,
                                     float* __restrict__ out) {
    const int wave  = threadIdx.x >> 5;
    const int lane  = threadIdx.x & 31;
    const int mtile = blockIdx.x;                 // 0..255
    const int ntile = blockIdx.y * 8 + wave;      // 0..15
    const int half  = lane >> 4;                  // 0 or 1
    const int l16   = lane & 15;
    const int koff  = half * 2;

    // Per-lane operand base pointers
    const float* arow = in + ((size_t)(mtile * 16 + l16)) * FIN; // A row (M = l16)
    const float* wcol = w + ntile * 16 + l16;                    // B col (N = l16), stride FOUT

    v8f acc = {};
#pragma unroll 4
    for (int k = 0; k < FIN; k += 4) {
        v2f a, b;
        a.x = arow[k + koff];
        a.y = arow[k + koff + 1];
        b.x = wcol[(size_t)(k + koff) * FOUT];
        b.y = wcol[(size_t)(k + koff + 1) * FOUT];
        // 8 args: (neg_a, A, neg_b, B, c_mod, C, reuse_a, reuse_b)
        acc = __builtin_amdgcn_wmma_f32_16x16x4_f32(
            false, a, false, b, (short)0, acc, false, false);
    }

    const int ncol = ntile * 16 + l16;
    const float bv = bias[ncol];
#pragma unroll
    for (int r = 0; r < 8; ++r) {
        const int mrow = mtile * 16 + r + half * 8; // global row in [0, B*N)
        const int i    = mrow & (N_ - 1);           // node index (N_ = 512 is pow2)
        const float rs = row_sum[mrow];
        const float h  = acc[r];
        float val = (rs * (1.0f / (float)N_) * h + rel_mean[i * FOUT + ncol])
                    / (rs + 1.0f) + bv;
        out[(size_t)mrow * FOUT + ncol] = fmaxf(val, 0.0f);
    }
}

// ---------------------------------------------------------------------------
// Launch
// inputs (setup_inputs order): input_tensor[B,N,FIN], adj[B,N,N],
//                              weight[FIN,FOUT], bias[FOUT], rel_pos[N,N,FOUT]
// output: float [B,N,FOUT]
// workspace: rel_mean (512*256 f32 = 512KB) | row_sum (4096 f32 = 16KB)
// ---------------------------------------------------------------------------
extern "C" void kernel_launch(void* const* d_in, const int* in_sizes, int n_in,
                              void* d_out, int out_size, void* d_ws, size_t ws_size,
                              hipStream_t stream) {
    const float* input  = (const float*)d_in[0];
    const float* adj    = (const float*)d_in[1];
    const float* weight = (const float*)d_in[2];
    const float* bias   = (const float*)d_in[3];
    const float* relpos = (const float*)d_in[4];
    float* out = (float*)d_out;

    float* rel_mean = (float*)d_ws;                       // 512*256 floats
    float* row_sum  = rel_mean + (size_t)N_ * FOUT;       // 4096 floats

    // 1) rel_mean: 512 blocks x 256 threads (streams 268MB with b128 loads)
    relmean_kernel<<<N_, FOUT, 0, stream>>>(relpos, rel_mean);

    // 2) row_sum: 4096 rows, 8 rows (waves) per block -> 512 blocks
    rowsum_kernel<<<MTOT / 8, 256, 0, stream>>>(adj, row_sum);

    // 3) WMMA GEMM + fused epilogue: (256, 2) blocks x 256 threads
    dim3 grid(MTOT / 16, FOUT / (16 * 8));
    gemm_epilogue_kernel<<<grid, 256, 0, stream>>>(input, weight, bias,
                                                   rel_mean, row_sum, out);
}